// Z_50259707298066
// MI455X (gfx1250) — compile-verified
//
#include <hip/hip_runtime.h>

// Pauli-Z phase kernel for DIM=2, S=1, wires {0,5,11} on an L-qubit state.
// out[n].re + i*out[n].im = (-1)^parity(n & mask) * (x_real[n] + i*x_imag[n])
// mask bits = (1 << (L-1-w)) for each target wire w.
//
// Pure HBM-streaming problem: 256 MB total traffic, ~11 us at 23.3 TB/s.
// Strategy: 128-bit coalesced loads/stores with gfx1250 non-temporal cache
// policy (stream > 192 MB L2, touched once), sign flip via sign-bit XOR.

typedef unsigned int u32;
typedef u32 __attribute__((ext_vector_type(4))) u32x4;

__global__ void __launch_bounds__(256)
z_phase_flip_kernel(const u32x4* __restrict__ xr,
                    const u32x4* __restrict__ xi,
                    u32x4* __restrict__ out,
                    u32 mask, u32 nvec) {
    u32 v = blockIdx.x * blockDim.x + threadIdx.x;
    if (v >= nvec) return;

    // First flat element index covered by this vector; mask bits are all
    // >= bit 12, so parity is uniform across the 4 consecutive elements.
    u32 n0   = v << 2;
    u32 sbit = (u32)(__popc(n0 & mask) & 1) << 31;  // 0 or 0x80000000

    // Non-temporal 128-bit loads (stream, no L2 reuse).
    u32x4 r = __builtin_nontemporal_load(xr + v);
    u32x4 m = __builtin_nontemporal_load(xi + v);

    // Conditional negation = XOR of IEEE-754 sign bit (exact, branch-free).
    r ^= sbit;
    m ^= sbit;

    // Interleave into complex64 layout: [r0,i0,r1,i1] [r2,i2,r3,i3].
    u32x4 o0 = { r.x, m.x, r.y, m.y };
    u32x4 o1 = { r.z, m.z, r.w, m.w };

    __builtin_nontemporal_store(o0, out + 2u * v);
    __builtin_nontemporal_store(o1, out + 2u * v + 1u);
}

extern "C" void kernel_launch(void* const* d_in, const int* in_sizes, int n_in,
                              void* d_out, int out_size, void* d_ws, size_t ws_size,
                              hipStream_t stream) {
    const u32x4* xr  = (const u32x4*)d_in[0];   // x_real: N float32
    const u32x4* xi  = (const u32x4*)d_in[1];   // x_imag: N float32
    u32x4*       out = (u32x4*)d_out;           // N complex64 -> 2N float32 interleaved

    const u32 N = (u32)in_sizes[0];             // 2^24 amplitudes

    // L = log2(N); wire w acts on bit (L-1-w) of the flat index.
    u32 L = 0;
    while ((1u << L) < N) ++L;
    const int wires[3] = {0, 5, 11};
    u32 mask = 0;
    for (int k = 0; k < 3; ++k) mask |= 1u << (L - 1u - (u32)wires[k]);

    const u32 nvec = N / 4u;                    // float4 vectors per input array
    const u32 block = 256u;
    const u32 grid  = (nvec + block - 1u) / block;

    z_phase_flip_kernel<<<grid, block, 0, stream>>>(xr, xi, out, mask, nvec);
}